// HC_MPNN_51685636440624
// MI455X (gfx1250) — compile-verified
//
#include <hip/hip_runtime.h>

#define D 64
#define NLAYERS 2
#define RREL 50
#define NNODES 20001
#define AR 6
#define NEDGE 30000
#define BATCH 4
#define NCAND 500
#define FEAT 128
#define LN_EPS 1e-5f

typedef __attribute__((ext_vector_type(2))) float v2f;
typedef __attribute__((ext_vector_type(8))) float v8f;

// ---------------------------------------------------------------- zero fill
__global__ void zero_kernel(float* __restrict__ p, long n) {
    long i = (long)blockIdx.x * blockDim.x + threadIdx.x;
    long stride = (long)gridDim.x * blockDim.x;
    for (; i < n; i += stride) p[i] = 0.0f;
}

// ------------------------------------------------- pos_search + query gather
__global__ void prep_kernel(const int* __restrict__ r_idx,
                            const int* __restrict__ ent,
                            const float* __restrict__ query_emb,
                            float* __restrict__ query,
                            int* __restrict__ pos_search) {
    int b = blockIdx.x;
    __shared__ int diff[AR];
    int t = threadIdx.x;            // blockDim = 256
    if (t < AR) diff[t] = 0;
    __syncthreads();
    for (int a = 0; a < AR; ++a) {
        int ref = ent[((long)b * NCAND + 0) * AR + a];
        for (int c = t; c < NCAND; c += blockDim.x) {
            if (ent[((long)b * NCAND + c) * AR + a] != ref) diff[a] = 1;  // benign race
        }
    }
    __syncthreads();
    if (t == 0) {
        int ps = 0;
        for (int a = 0; a < AR; ++a) { if (diff[a]) { ps = a; break; } }
        pos_search[b] = ps;         // argmax(~mask): first differing position, else 0
    }
    if (t < D) query[b * D + t] = query_emb[r_idx[(long)b * NCAND + 0] * D + t];
}

// ------------------------------------------------------- init node features
__global__ void init_scatter_kernel(float* __restrict__ h,
                                    const float* __restrict__ query,
                                    const int* __restrict__ pos_search,
                                    const int* __restrict__ ent,
                                    const int* __restrict__ arity,
                                    const float* __restrict__ pos_table) {
    int b = blockIdx.x;
    int d = threadIdx.x;            // 64 threads, each owns one feature column
    int ps = pos_search[b];
    int ar = arity[(long)b * NCAND + 0];
    float q = query[b * D + d];
    for (int a = 0; a < AR; ++a) {  // serial over a: duplicate nodes accumulate race-free
        if (a < ar && a != ps) {
            int n = ent[((long)b * NCAND + 0) * AR + a];
            if (n != 0)
                h[((long)(b * NNODES + n)) * D + d] += q + pos_table[(a + 1) * D + d];
        }
    }
}

// ------------------------------------------------ per-edge messages + scatter
__global__ void message_kernel(const float* __restrict__ h,
                               float* __restrict__ agg,
                               const int* __restrict__ edge_list,
                               const int* __restrict__ rel_list,
                               const float* __restrict__ rel_emb_l,
                               const float* __restrict__ pos_table) {
    int e = blockIdx.x;
    int d = threadIdx.x;            // 0..63 feature
    int b = threadIdx.y;            // 0..3 batch
    int rel = rel_list[e];
    float r = rel_emb_l[rel * D + d];
    float m[AR];
    int nodes[AR];
    float sum = 0.0f;
#pragma unroll
    for (int a = 0; a < AR; ++a) {
        int n = edge_list[(long)e * AR + a];
        nodes[a] = n;
        float mv = 0.0f;
        if (n != 0) {               // invalid slots contribute exactly 0 in the reference
            mv = (h[((long)(b * NNODES + n)) * D + d] + pos_table[(a + 1) * D + d]) * r;
            sum += mv;
        }
        m[a] = mv;
    }
#pragma unroll
    for (int a = 0; a < AR; ++a) {
        if (nodes[a] != 0) {
            atomicAdd(&agg[((long)(b * NNODES + nodes[a])) * D + d], sum - m[a]);
        }
    }
}

// ---------------- node update: [agg|h] (Mx128) @ [W_agg;W_self] (128x64) via
// ---------------- V_WMMA_F32_16X16X4_F32, then bias + LayerNorm + ReLU + res
__global__ void __launch_bounds__(128)
node_update_kernel(float* __restrict__ h, const float* __restrict__ agg,
                   const float* __restrict__ Wagg, const float* __restrict__ Wself,
                   const float* __restrict__ blin, const float* __restrict__ lng,
                   const float* __restrict__ lnb, int Mtot) {
    __shared__ float tile[16 * D];       // 16 rows x 64 cols GEMM result
    __shared__ float red_s[4][16];
    __shared__ float red_q[4][16];
    __shared__ float stats[16][2];       // mu, 1/sigma

    int tid  = threadIdx.x;
    int wave = tid >> 5;                 // 4 waves; wave owns cols [16w,16w+16)
    int lane = tid & 31;
    int row0 = blockIdx.x * 16;
    int nb   = wave * 16;

    int mrow = row0 + (lane & 15);
    if (mrow >= Mtot) mrow = Mtot - 1;   // clamp: keep EXEC all-ones for WMMA
    int khalf = (lane >> 4) << 1;        // lanes 0-15 -> K {0,1}; 16-31 -> K {2,3}
    int col   = nb + (lane & 15);

    const float* xa = agg + (long)mrow * D;
    const float* xs = h   + (long)mrow * D;

    v8f c = {0.f, 0.f, 0.f, 0.f, 0.f, 0.f, 0.f, 0.f};
#pragma unroll
    for (int kb = 0; kb < D; kb += 4) {  // agg @ W_agg
        int k = kb + khalf;
        v2f a, bm;
        a.x  = xa[k];               a.y  = xa[k + 1];
        bm.x = Wagg[k * D + col];   bm.y = Wagg[(k + 1) * D + col];
        c = __builtin_amdgcn_wmma_f32_16x16x4_f32(false, a, false, bm,
                                                  (short)0, c, false, false);
    }
#pragma unroll
    for (int kb = 0; kb < D; kb += 4) {  // h @ W_self (same accumulator)
        int k = kb + khalf;
        v2f a, bm;
        a.x  = xs[k];               a.y  = xs[k + 1];
        bm.x = Wself[k * D + col];  bm.y = Wself[(k + 1) * D + col];
        c = __builtin_amdgcn_wmma_f32_16x16x4_f32(false, a, false, bm,
                                                  (short)0, c, false, false);
    }

    // C/D layout: VGPR i of lane L -> element (M = i + 8*(L>=16), N = L%16)
#pragma unroll
    for (int i = 0; i < 8; ++i)
        tile[(i + ((lane >> 4) << 3)) * D + nb + (lane & 15)] = c[i];
    __syncthreads();

    // LayerNorm partial sums: 64 threads = 16 rows x 4 segments of 16 cols
    if (tid < 64) {
        int r = tid & 15, seg = tid >> 4;
        float s = 0.f, q = 0.f;
#pragma unroll
        for (int j = 0; j < 16; ++j) {
            int cc = seg * 16 + j;
            float v = tile[r * D + cc] + blin[cc];
            s += v; q += v * v;
        }
        red_s[seg][r] = s; red_q[seg][r] = q;
    }
    __syncthreads();
    if (tid < 16) {
        float s = red_s[0][tid] + red_s[1][tid] + red_s[2][tid] + red_s[3][tid];
        float q = red_q[0][tid] + red_q[1][tid] + red_q[2][tid] + red_q[3][tid];
        float mu  = s * (1.0f / 64.0f);
        float var = fmaxf(q * (1.0f / 64.0f) - mu * mu, 0.0f);
        stats[tid][0] = mu;
        stats[tid][1] = rsqrtf(var + LN_EPS);
    }
    __syncthreads();

    // normalize + ReLU + residual, in place (each row owned by this block only)
#pragma unroll
    for (int i = 0; i < 8; ++i) {
        int idx = tid * 8 + i;
        int r = idx >> 6;
        int cc = idx & 63;
        int grow = row0 + r;
        if (grow < Mtot) {
            float v = tile[r * D + cc] + blin[cc];
            float y = (v - stats[r][0]) * stats[r][1] * lng[cc] + lnb[cc];
            h[(long)grow * D + cc] += fmaxf(y, 0.0f);
        }
    }
}

// --------------------------------------------------------- final MLP scoring
__global__ void __launch_bounds__(128)
score_kernel(const float* __restrict__ h, const float* __restrict__ query,
             const int* __restrict__ pos_search, const int* __restrict__ ent,
             const float* __restrict__ W1, const float* __restrict__ b1,
             const float* __restrict__ W2, const float* __restrict__ b2,
             float* __restrict__ out) {
    int bc = blockIdx.x;
    int b = bc / NCAND, c = bc % NCAND;
    __shared__ float feat[FEAT];
    __shared__ float red[FEAT];
    int t = threadIdx.x;                 // 128
    int ps = pos_search[b];
    int node = ent[((long)b * NCAND + c) * AR + ps];
    feat[t] = (t < D) ? h[((long)(b * NNODES + node)) * D + t]
                      : query[b * D + (t - D)];
    __syncthreads();
    float acc = b1[t];
#pragma unroll 8
    for (int k = 0; k < FEAT; ++k) acc += feat[k] * W1[k * FEAT + t];
    red[t] = fmaxf(acc, 0.0f) * W2[t];
    __syncthreads();
    for (int s = 64; s > 0; s >>= 1) {
        if (t < s) red[t] += red[t + s];
        __syncthreads();
    }
    if (t == 0) out[bc] = red[0] + b2[0];
}

// ---------------------------------------------------------------------------
extern "C" void kernel_launch(void* const* d_in, const int* in_sizes, int n_in,
                              void* d_out, int out_size, void* d_ws, size_t ws_size,
                              hipStream_t stream) {
    const int*   r_idx   = (const int*)d_in[0];
    const int*   ent     = (const int*)d_in[1];
    const int*   arity   = (const int*)d_in[2];
    const int*   edge_l  = (const int*)d_in[3];
    const int*   rel_l   = (const int*)d_in[4];
    const float* pos_tab = (const float*)d_in[5];
    const float* q_emb   = (const float*)d_in[6];
    const float* rel_emb = (const float*)d_in[7];
    const float* W_agg   = (const float*)d_in[8];
    const float* W_self  = (const float*)d_in[9];
    const float* b_lin   = (const float*)d_in[10];
    const float* ln_g    = (const float*)d_in[11];
    const float* ln_b    = (const float*)d_in[12];
    const float* W1      = (const float*)d_in[13];
    const float* b1      = (const float*)d_in[14];
    const float* W2      = (const float*)d_in[15];
    const float* b2      = (const float*)d_in[16];

    char* ws = (char*)d_ws;
    const long nodeElems = (long)BATCH * NNODES * D;        // 5,120,256 floats
    float* h_buf   = (float*)ws;
    float* agg_buf = (float*)(ws + nodeElems * sizeof(float));
    float* query   = (float*)(ws + 2 * nodeElems * sizeof(float));
    int*   ps_buf  = (int*)  (ws + 2 * nodeElems * sizeof(float) + 1024);

    // zero h and agg in one pass (they are adjacent)
    zero_kernel<<<2048, 256, 0, stream>>>(h_buf, 2 * nodeElems);
    prep_kernel<<<BATCH, 256, 0, stream>>>(r_idx, ent, q_emb, query, ps_buf);
    init_scatter_kernel<<<BATCH, D, 0, stream>>>(h_buf, query, ps_buf, ent, arity, pos_tab);

    const int Mtot = BATCH * NNODES;
    const int gemmBlocks = (Mtot + 15) / 16;
    dim3 mblock(D, BATCH);
    for (int l = 0; l < NLAYERS; ++l) {
        if (l > 0) zero_kernel<<<2048, 256, 0, stream>>>(agg_buf, nodeElems);
        message_kernel<<<NEDGE, mblock, 0, stream>>>(h_buf, agg_buf, edge_l, rel_l,
                                                     rel_emb + (long)l * RREL * D, pos_tab);
        node_update_kernel<<<gemmBlocks, 128, 0, stream>>>(
            h_buf, agg_buf,
            W_agg + (long)l * D * D, W_self + (long)l * D * D,
            b_lin + l * D, ln_g + l * D, ln_b + l * D, Mtot);
    }
    score_kernel<<<BATCH * NCAND, 128, 0, stream>>>(h_buf, query, ps_buf, ent,
                                                    W1, b1, W2, b2, (float*)d_out);
}